// GlobalAttention_78400333021290
// MI455X (gfx1250) — compile-verified
//
#include <hip/hip_runtime.h>
#include <math.h>

typedef __attribute__((ext_vector_type(2))) float v2f;
typedef __attribute__((ext_vector_type(8))) float v8f;
typedef __attribute__((ext_vector_type(4))) unsigned int v4u;
typedef __attribute__((ext_vector_type(8))) unsigned int v8u;

#define DD 1024
#define SS 4096
#define BB 32
#define CHUNK 64
#define NCHUNK (SS / CHUNK) // 64

// ---------------------------------------------------------------------------
// Kernel 1: q_proj[32,1024] = query[32,1024] @ Wa[1024,1024] + bias
// One wave per 16x16 output tile, V_WMMA_F32_16X16X4_F32, K stepped by 4.
// Tiles: 2 (M) x 64 (N) = 128 waves = 32 blocks x 4 waves.
// ---------------------------------------------------------------------------
__global__ __launch_bounds__(128)
void ga_qproj_wmma(const float* __restrict__ q,
                   const float* __restrict__ w,
                   const float* __restrict__ bias,
                   float* __restrict__ qp) {
  int wave = (blockIdx.x * blockDim.x + threadIdx.x) >> 5;
  int lane = threadIdx.x & 31;
  int mt = wave & 1;        // 2 M-tiles (B=32)
  int nt = wave >> 1;       // 64 N-tiles
  int mbase = mt * 16, nbase = nt * 16;
  int half = lane >> 4, lid = lane & 15;

  v8f c = {};
  const float* qrow = q + (size_t)(mbase + lid) * DD;
  for (int k = 0; k < DD; k += 4) {
    int kk = k + 2 * half;          // lanes 0-15: K+0/K+1 ; lanes 16-31: K+2/K+3
    v2f a, b;
    a.x = qrow[kk];
    a.y = qrow[kk + 1];
    b.x = w[(size_t)kk * DD + nbase + lid];
    b.y = w[(size_t)(kk + 1) * DD + nbase + lid];
    c = __builtin_amdgcn_wmma_f32_16x16x4_f32(false, a, false, b,
                                              (short)0, c, false, false);
  }
  float bv = bias[nbase + lid];
#pragma unroll
  for (int r = 0; r < 8; ++r) {
    int row = mbase + r + 8 * half; // C layout: VGPR r = M=r (lanes 0-15) / M=r+8
    qp[(size_t)row * DD + nbase + lid] = c[r] + bv;
  }
}

// ---------------------------------------------------------------------------
// Kernel 2: per (b, 64-row chunk): TDM-stage values chunk into LDS (256 KB,
// one tensor_load_to_lds DMA), compute scores, chunk softmax stats (m,l),
// partial context. values is read ONCE from HBM.
// grid = 32*64 blocks, 256 threads (8 waves).
// ---------------------------------------------------------------------------
__global__ __launch_bounds__(256)
void ga_chunk_pass(const float* __restrict__ values,
                   const float* __restrict__ qp,
                   float* __restrict__ ws_scores,  // [32,4096]
                   float* __restrict__ ws_ml,      // [32,64,2]
                   float* __restrict__ ws_ctx) {   // [32,64,1024]
  __shared__ float4 v4[CHUNK * DD / 4];  // 256 KB (CDNA5: 320 KB/WGP)
  __shared__ float4 qp4[DD / 4];         // 4 KB
  __shared__ float sc[CHUNK];
  __shared__ float wv[CHUNK];

  int b = blockIdx.x >> 6;
  int chunk = blockIdx.x & 63;
  int s0 = chunk * CHUNK;
  int tid = threadIdx.x;
  int wave = tid >> 5, lane = tid & 31;

  // ---- Tensor Data Mover: DMA 64x1024 fp32 tile Global -> LDS ----
  // Issued by wave 0 only (wave-uniform branch: tensor ops ignore EXEC,
  // so other waves must branch around the instruction entirely).
  if (wave == 0) {
    unsigned int lds_base = (unsigned int)(uintptr_t)(&v4[0]);
    unsigned long long ga =
        (unsigned long long)(uintptr_t)(values + (size_t)(b * SS + s0) * DD);
    // D# group 0: [1:0]=count=1 | [63:32]=lds_addr | [120:64]=global_addr |
    //             [127:126]=type=2
    v4u g0;
    g0[0] = 1u;
    g0[1] = lds_base;
    g0[2] = (unsigned int)(ga & 0xFFFFFFFFull);
    g0[3] = (unsigned int)((ga >> 32) & 0x1FFFFFFull) | (2u << 30);
    // D# group 1: data_size=2 (4B) | tensor_dim0=1024 | tensor_dim1=64 |
    //             tile_dim0=1024 | tile_dim1=64 | tensor_dim0_stride=1024
    v8u g1;
    g1[0] = 2u << 16;          // data_size = 4 bytes; mask/flags/pad = 0
    g1[1] = 1024u << 16;       // tensor_dim0[15:0]  (bits 79:48)
    g1[2] = 64u << 16;         // tensor_dim1[15:0]  (bits 111:80)
    g1[3] = 1024u << 16;       // tile_dim0          (bits 127:112)
    g1[4] = 64u;               // tile_dim1          (bits 143:128)
    g1[5] = 1024u;             // tensor_dim0_stride (bits 207:160)
    g1[6] = 0u;
    g1[7] = 0u;
    asm volatile("tensor_load_to_lds %0, %1" :: "s"(g0), "s"(g1) : "memory");
  }

  // Stage q_proj row into LDS with regular loads (overlaps the TDM DMA)
  const float4* qg = (const float4*)(qp + (size_t)b * DD);
  qp4[tid] = qg[tid];  // 256 threads x float4 = 1024 floats

  if (wave == 0) {
    __builtin_amdgcn_s_wait_tensorcnt(0);  // TDM tile landed in LDS
  }
  __syncthreads();

  // Scores: each wave handles 8 rows; wave-dot over D from LDS
  for (int r = 0; r < 8; ++r) {
    int s = wave * 8 + r;
    float acc = 0.f;
#pragma unroll
    for (int j = 0; j < 8; ++j) {
      float4 a = v4[s * (DD / 4) + j * 32 + lane];
      float4 qv = qp4[j * 32 + lane];
      acc += a.x * qv.x + a.y * qv.y + a.z * qv.z + a.w * qv.w;
    }
#pragma unroll
    for (int off = 16; off > 0; off >>= 1) acc += __shfl_xor(acc, off, 32);
    if (lane == 0) sc[s] = acc;
  }
  __syncthreads();

  // Chunk softmax stats (wave 0): m = max, l = sum exp(s - m); wv = exp(s - m)
  if (wave == 0) {
    float a0 = sc[lane], a1 = sc[lane + 32];
    float mx = fmaxf(a0, a1);
#pragma unroll
    for (int off = 16; off > 0; off >>= 1) mx = fmaxf(mx, __shfl_xor(mx, off, 32));
    float e0 = __expf(a0 - mx), e1 = __expf(a1 - mx);
    wv[lane] = e0;
    wv[lane + 32] = e1;
    float ls = e0 + e1;
#pragma unroll
    for (int off = 16; off > 0; off >>= 1) ls += __shfl_xor(ls, off, 32);
    if (lane == 0) {
      ws_ml[(b * NCHUNK + chunk) * 2 + 0] = mx;
      ws_ml[(b * NCHUNK + chunk) * 2 + 1] = ls;
    }
  }
  if (tid < CHUNK) ws_scores[b * SS + s0 + tid] = sc[tid];
  __syncthreads();

  // Partial context from LDS: thread owns 4 output dims
  float4 acc = {0.f, 0.f, 0.f, 0.f};
  for (int s = 0; s < CHUNK; ++s) {
    float wgt = wv[s];
    float4 v = v4[s * (DD / 4) + tid];
    acc.x += wgt * v.x; acc.y += wgt * v.y;
    acc.z += wgt * v.z; acc.w += wgt * v.w;
  }
  float4* ctx4 = (float4*)(ws_ctx + (size_t)(b * NCHUNK + chunk) * DD);
  ctx4[tid] = acc;
}

// ---------------------------------------------------------------------------
// Kernel 3: per-batch combine across 64 chunks (online-softmax rescale),
// write context to d_out[0 : 32*1024], save global (m, L).
// ---------------------------------------------------------------------------
__global__ __launch_bounds__(256)
void ga_combine(const float* __restrict__ ws_ml,
                const float* __restrict__ ws_ctx,
                float* __restrict__ out_ctx,
                float* __restrict__ ws_mlf) {
  __shared__ float s_m, s_l;
  __shared__ float sf[NCHUNK];
  int b = blockIdx.x;
  int tid = threadIdx.x;

  if (tid == 0) {
    float m = -INFINITY;
    for (int i = 0; i < NCHUNK; ++i) m = fmaxf(m, ws_ml[(b * NCHUNK + i) * 2]);
    float l = 0.f;
    for (int i = 0; i < NCHUNK; ++i)
      l += ws_ml[(b * NCHUNK + i) * 2 + 1] * __expf(ws_ml[(b * NCHUNK + i) * 2] - m);
    s_m = m;
    s_l = l;
    ws_mlf[b * 2 + 0] = m;
    ws_mlf[b * 2 + 1] = l;
  }
  __syncthreads();
  if (tid < NCHUNK) sf[tid] = __expf(ws_ml[(b * NCHUNK + tid) * 2] - s_m) / s_l;
  __syncthreads();

  float4 acc = {0.f, 0.f, 0.f, 0.f};
  for (int i = 0; i < NCHUNK; ++i) {
    const float4* c4 = (const float4*)(ws_ctx + (size_t)(b * NCHUNK + i) * DD);
    float4 v = c4[tid];
    float f = sf[i];
    acc.x += f * v.x; acc.y += f * v.y;
    acc.z += f * v.z; acc.w += f * v.w;
  }
  ((float4*)(out_ctx + (size_t)b * DD))[tid] = acc;
}

// ---------------------------------------------------------------------------
// Kernel 4: attention weights = exp(score - m) / L -> d_out[32768 : 163840]
// ---------------------------------------------------------------------------
__global__ __launch_bounds__(256)
void ga_weights(const float* __restrict__ ws_scores,
                const float* __restrict__ ws_mlf,
                float* __restrict__ out_w) {
  int idx = blockIdx.x * blockDim.x + threadIdx.x;
  int b = idx >> 12;
  out_w[idx] = __expf(ws_scores[idx] - ws_mlf[b * 2]) / ws_mlf[b * 2 + 1];
}

// ---------------------------------------------------------------------------
extern "C" void kernel_launch(void* const* d_in, const int* in_sizes, int n_in,
                              void* d_out, int out_size, void* d_ws, size_t ws_size,
                              hipStream_t stream) {
  const float* query  = (const float*)d_in[0];  // [32,1024]
  const float* values = (const float*)d_in[1];  // [32,4096,1024]
  const float* Wa     = (const float*)d_in[2];  // [1024,1024]
  const float* bias   = (const float*)d_in[3];  // [1024]

  float* out_ctx = (float*)d_out;                 // [32,1024]
  float* out_w   = (float*)d_out + BB * DD;       // [32,4096]

  // Workspace layout (floats)
  float* ws       = (float*)d_ws;
  float* ws_qp    = ws;                           // 32*1024      = 32768
  float* ws_sc    = ws_qp + BB * DD;              // 32*4096      = 131072
  float* ws_ml    = ws_sc + BB * SS;              // 32*64*2      = 4096
  float* ws_mlf   = ws_ml + BB * NCHUNK * 2;      // 64
  float* ws_ctx   = ws_mlf + 64;                  // 32*64*1024   = 2097152  (16B aligned)

  ga_qproj_wmma<<<32, 128, 0, stream>>>(query, Wa, bias, ws_qp);
  ga_chunk_pass<<<BB * NCHUNK, 256, 0, stream>>>(values, ws_qp, ws_sc, ws_ml, ws_ctx);
  ga_combine<<<BB, 256, 0, stream>>>(ws_ml, ws_ctx, out_ctx, ws_mlf);
  ga_weights<<<(BB * SS) / 256, 256, 0, stream>>>(ws_sc, ws_mlf, out_w);
}